// MultiScaleExpertCompanion_26104811225654
// MI455X (gfx1250) — compile-verified
//
#include <hip/hip_runtime.h>
#include <hip/hip_bf16.h>
#include <math.h>

// Problem constants (match reference)
#define B_    2
#define S_    2048
#define DIM_  512
#define H_    8
#define HD_   64
#define KK_   64            // routes per query
#define M_    (B_ * S_)     // 4096 rows
#define N1_   (3 * DIM_)    // 1536 qkv columns
#define SCALE_ 0.125f       // 1/sqrt(64)

#define KC_      16         // K-chunk staged in LDS per iteration
#define WSTRIDE_ 72         // padded LDS row stride (banks: disjoint half-wave groups)

typedef __attribute__((ext_vector_type(2))) float v2f;
typedef __attribute__((ext_vector_type(8))) float v8f;
typedef int v4i __attribute__((vector_size(16)));   // matches builtin param type

// ---- CDNA5 async global->LDS copy (ASYNCcnt-tracked) ---------------------
__device__ __forceinline__ void async_copy_b128(const void* gsrc, void* lds_dst) {
#if __has_builtin(__builtin_amdgcn_global_load_async_to_lds_b128)
    __builtin_amdgcn_global_load_async_to_lds_b128(
        (v4i*)gsrc, (v4i*)lds_dst, 0, 0);
#else
    unsigned loff = (unsigned)(size_t)(__attribute__((address_space(3))) void*)lds_dst;
    asm volatile("global_load_async_to_lds_b128 %0, %1, off"
                 :: "v"(loff), "v"(gsrc) : "memory");
#endif
}

__device__ __forceinline__ void wait_async0() {
#if __has_builtin(__builtin_amdgcn_s_wait_asynccnt)
    __builtin_amdgcn_s_wait_asynccnt(0);
#else
    asm volatile("s_wait_asynccnt 0x0" ::: "memory");
#endif
}

// -------------------------------------------------------------------------
// C[M,N] = A[M,Kd] @ W[Kd,N] + bias[N]    (fp32, WMMA f32 16x16x4)
//
// Block = 256 threads = 8 waves; block tile 128 x 64 of C (wave: 16 x 64).
// Per KC_=16 K-chunk: W[16 x 64] staged in LDS via async b128 copies
// (double-buffered), B fragments then read from LDS; A fragments are
// per-wave b64 global loads.
//
// f32 WMMA operand layouts (ISA 7.12.2):
//  A (16x4): lanes 0-15 -> M=lane,     VGPR0=K+0, VGPR1=K+1
//            lanes 16-31-> M=lane-16,  VGPR0=K+2, VGPR1=K+3
//  B (4x16): lanes 0-15 -> N=lane,     VGPR0=row K+0, VGPR1=row K+1
//            lanes 16-31-> N=lane-16,  VGPR0=row K+2, VGPR1=row K+3
//  C/D (16x16): VGPR i: lanes 0-15 -> M=i, lanes 16-31 -> M=i+8, N=lane%16
// -------------------------------------------------------------------------
__global__ __launch_bounds__(256) void gemm_bias_f32_wmma(
    const float* __restrict__ A, const float* __restrict__ W,
    const float* __restrict__ bias, float* __restrict__ C,
    int M, int N, int Kd)
{
    __shared__ float wbuf[2][KC_ * WSTRIDE_];   // 2 x 4.5 KB

    const unsigned tid  = threadIdx.x;
    const unsigned lane = tid & 31u;
    const unsigned wave = tid >> 5;
    const int tileM = blockIdx.y * 128 + (int)wave * 16;
    const int tileN = blockIdx.x * 64;
    const unsigned l15 = lane & 15u;
    const bool hi = lane >= 16u;

    // cooperative-staging role: thread t -> row (t>>4) of chunk, cols (t&15)*4 .. +3
    const int wr = (int)(tid >> 4);        // 0..15
    const int wc = (int)(tid & 15u) * 4;   // 0,4,...,60

    // issue chunk 0
    async_copy_b128(W + (size_t)wr * N + tileN + wc, &wbuf[0][wr * WSTRIDE_ + wc]);

    v8f acc0 = {}, acc1 = {}, acc2 = {}, acc3 = {};
    const float* Arow = A + (size_t)(tileM + (int)l15) * Kd;

    const int nch = Kd / KC_;
    for (int ch = 0; ch < nch; ++ch) {
        wait_async0();          // chunk ch resident in wbuf[ch&1]
        __syncthreads();        // all waves done with wbuf[(ch+1)&1] reads (iter ch-1)

        if (ch + 1 < nch) {
            const float* src = W + (size_t)((ch + 1) * KC_ + wr) * N + tileN + wc;
            async_copy_b128(src, &wbuf[(ch + 1) & 1][wr * WSTRIDE_ + wc]);
        }

        const float* buf = wbuf[ch & 1];
        const int kb = ch * KC_;
#pragma unroll
        for (int ks = 0; ks < KC_; ks += 4) {
            v2f a = *(const v2f*)(Arow + kb + ks + (hi ? 2 : 0));

            const int rr = ks + (hi ? 2 : 0);
            const float* L0 = buf + rr * WSTRIDE_ + l15;
            const float* L1 = L0 + WSTRIDE_;

            v2f b0, b1, b2, b3;
            b0.x = L0[0];  b0.y = L1[0];
            b1.x = L0[16]; b1.y = L1[16];
            b2.x = L0[32]; b2.y = L1[32];
            b3.x = L0[48]; b3.y = L1[48];

            acc0 = __builtin_amdgcn_wmma_f32_16x16x4_f32(false, a, false, b0, (short)0, acc0, false, false);
            acc1 = __builtin_amdgcn_wmma_f32_16x16x4_f32(false, a, false, b1, (short)0, acc1, false, false);
            acc2 = __builtin_amdgcn_wmma_f32_16x16x4_f32(false, a, false, b2, (short)0, acc2, false, false);
            acc3 = __builtin_amdgcn_wmma_f32_16x16x4_f32(false, a, false, b3, (short)0, acc3, false, false);
        }
    }

    const int rbase = tileM + (hi ? 8 : 0);
    const float bz0 = bias[tileN + (int)l15];
    const float bz1 = bias[tileN + 16 + (int)l15];
    const float bz2 = bias[tileN + 32 + (int)l15];
    const float bz3 = bias[tileN + 48 + (int)l15];
#pragma unroll
    for (int i = 0; i < 8; ++i) {
        float* Crow = C + (size_t)(rbase + i) * N + tileN + l15;
        Crow[0]  = acc0[i] + bz0;
        Crow[16] = acc1[i] + bz1;
        Crow[32] = acc2[i] + bz2;
        Crow[48] = acc3[i] + bz3;
    }
}

// -------------------------------------------------------------------------
// Routed-gather attention core. One 64-thread block per (b, h, q).
// qkv layout: row (b*S + s) of 1536 cols; q at h*64, k at 512+h*64, v at 1024+h*64.
// Phase 1: thread t = key t -> score; LDS softmax across 64 keys.
// Phase 2: thread t = output dim d -> sum_j p_j * V[route_j][d] (coalesced).
// attn_out stored as [B, S, H*HD] so the output projection reads it row-major.
// -------------------------------------------------------------------------
__global__ __launch_bounds__(64) void route_attn(
    const float* __restrict__ qkv, const int* __restrict__ routes,
    float* __restrict__ attn_out)
{
    const int q = blockIdx.x % S_;
    const int h = (blockIdx.x / S_) % H_;
    const int b = blockIdx.x / (S_ * H_);
    const int t = threadIdx.x;      // 0..63

    __shared__ float qs[HD_];
    __shared__ float sc[KK_];
    __shared__ int   kix[KK_];

    const size_t rowq = (size_t)(b * S_ + q) * N1_;
    qs[t] = qkv[rowq + h * HD_ + t];
    const int kidx = routes[q * KK_ + t];
    kix[t] = kidx;
    __syncthreads();

    // score for key t
    const float* krow = qkv + (size_t)(b * S_ + kidx) * N1_ + DIM_ + h * HD_;
    float s = 0.f;
#pragma unroll 8
    for (int d = 0; d < HD_; ++d) s = fmaf(qs[d], krow[d], s);
    s *= SCALE_;
    sc[t] = s;
    __syncthreads();

    float m = -INFINITY;
#pragma unroll 8
    for (int j = 0; j < KK_; ++j) m = fmaxf(m, sc[j]);
    __syncthreads();

    const float e = __expf(s - m);
    sc[t] = e;
    __syncthreads();

    float denom = 0.f;
#pragma unroll 8
    for (int j = 0; j < KK_; ++j) denom += sc[j];
    const float inv = 1.f / denom;

    // output dim t
    float acc = 0.f;
    for (int j = 0; j < KK_; ++j) {
        const float* vrow = qkv + (size_t)(b * S_ + kix[j]) * N1_ + 2 * DIM_ + h * HD_;
        acc = fmaf(sc[j], vrow[t], acc);
    }
    attn_out[(size_t)(b * S_ + q) * DIM_ + h * HD_ + t] = acc * inv;
}

// -------------------------------------------------------------------------
extern "C" void kernel_launch(void* const* d_in, const int* in_sizes, int n_in,
                              void* d_out, int out_size, void* d_ws, size_t ws_size,
                              hipStream_t stream) {
    (void)in_sizes; (void)n_in; (void)out_size; (void)ws_size;

    const float* x      = (const float*)d_in[0];   // [B,S,DIM]
    const float* qkv_w  = (const float*)d_in[1];   // [DIM, 3*DIM]
    const float* qkv_b  = (const float*)d_in[2];   // [3*DIM]
    const float* out_w  = (const float*)d_in[3];   // [DIM, DIM]
    const float* out_b  = (const float*)d_in[4];   // [DIM]
    const int*   routes = (const int*)d_in[5];     // [S, K]
    float*       out    = (float*)d_out;           // [B,S,DIM]

    float* qkv  = (float*)d_ws;                    // [M, 1536] = 24 MB
    float* attn = qkv + (size_t)M_ * N1_;          // [M, 512]  =  8 MB

    // 1) QKV projection: [4096,512] @ [512,1536] + b
    {
        dim3 grid(N1_ / 64, M_ / 128);
        hipLaunchKernelGGL(gemm_bias_f32_wmma, grid, dim3(256), 0, stream,
                           x, qkv_w, qkv_b, qkv, M_, N1_, DIM_);
    }
    // 2) Routed attention core
    {
        hipLaunchKernelGGL(route_attn, dim3(B_ * H_ * S_), dim3(64), 0, stream,
                           qkv, routes, attn);
    }
    // 3) Output projection: [4096,512] @ [512,512] + b -> d_out
    {
        dim3 grid(DIM_ / 64, M_ / 128);
        hipLaunchKernelGGL(gemm_bias_f32_wmma, grid, dim3(256), 0, stream,
                           attn, out_w, out_b, out, M_, DIM_, DIM_);
    }
}